// Contamination_21577915695283
// MI455X (gfx1250) — compile-verified
//
#include <hip/hip_runtime.h>
#include <stdint.h>

// ----------------------------------------------------------------------------
// Contamination stencil: y = x + 0.2 * (sum of 8 zero-padded neighbors)
// Shape: (8,16,1024,1024) f32 -> 128 planes of 1024x1024. Memory-bound:
// ~1 GiB traffic vs ~1.2 GFLOP => target is stream bandwidth (~46 us floor).
// CDNA5 paths: TDM tensor_load_to_lds (TENSORcnt) for interior tiles,
// async global->LDS (ASYNCcnt) for edge tiles, LDS compute, NT b128 stores.
// ----------------------------------------------------------------------------

#define HH 1024
#define WW 1024
#define NPLANES 128          // B*C
#define TX 128               // outputs per block in x
#define TY 32                // outputs per block in y
#define RROWS (TY + 2)       // 34 staged rows   (gy = y0-1 .. y0+TY)
#define RCOLS 136            // staged cols      (gx = x0-4 .. x0+131)
#define LSTRIDE 136          // LDS row stride == RCOLS (TDM writes contiguous)
#define ALPHA 0.2f

typedef __attribute__((ext_vector_type(4))) float f32x4;
typedef __attribute__((ext_vector_type(4))) int   i32x4;
typedef __attribute__((ext_vector_type(8))) int   i32x8;

// Low 32 bits of a generic pointer to __shared__ == wave-relative LDS byte addr
// (ISA 10.2).
__device__ __forceinline__ uint32_t lds_byte_addr(const void* p) {
  return (uint32_t)(uintptr_t)p;
}

__device__ __forceinline__ void async_load_b32(uint32_t lds_dst, const float* gsrc) {
  asm volatile("global_load_async_to_lds_b32 %0, %1, off"
               :: "v"(lds_dst), "v"((uint64_t)(uintptr_t)gsrc)
               : "memory");
}

__device__ __forceinline__ void wait_asynccnt0() {
#if __has_builtin(__builtin_amdgcn_s_wait_asynccnt)
  __builtin_amdgcn_s_wait_asynccnt(0);
#else
  asm volatile("s_wait_asynccnt 0" ::: "memory");
#endif
}

__device__ __forceinline__ void wait_tensorcnt0() {
#if __has_builtin(__builtin_amdgcn_s_wait_tensorcnt)
  __builtin_amdgcn_s_wait_tensorcnt(0);
#else
  asm volatile("s_wait_tensorcnt 0" ::: "memory");
#endif
}

// One TDM DMA: 2D tile (RROWS x RCOLS f32, global row stride WW) -> LDS.
// D# layout per CDNA5 ISA (groups 0 and 1; 2D tensor,
// VADDR2/3 omitted). All fields are block-uniform -> SGPR tuples.
__device__ __forceinline__ void tdm_load_tile(uint32_t lds_dst, const float* gsrc) {
  const uint64_t ga = (uint64_t)(uintptr_t)gsrc;

  i32x4 g0;
  g0[0] = 1;                                   // count=1, is_restore=0, no gather
  g0[1] = (int)lds_dst;                        // lds_addr[31:0]
  g0[2] = (int)(uint32_t)ga;                   // global_addr[31:0]
  g0[3] = (int)((uint32_t)(ga >> 32) | 0x80000000u); // addr[56:32] | type=2<<30

  i32x8 g1;
  g1[0] = 0x00020000;                          // wg_mask=0, data_size=2 (4B)
  g1[1] = (RCOLS << 16);                       // atomic_barrier=0 | tensor_dim0 lo16
  g1[2] = (RROWS << 16);                       // tensor_dim0 hi16=0 | tensor_dim1 lo16
  g1[3] = (RCOLS << 16);                       // tensor_dim1 hi16=0 | tile_dim0
  g1[4] = RROWS;                               // tile_dim1 | tile_dim2=0
  g1[5] = WW;                                  // tensor_dim0_stride[31:0] = 1024
  g1[6] = 0;                                   // stride hi | tensor_dim1_stride lo
  g1[7] = 0;                                   // tensor_dim1_stride hi

  asm volatile("tensor_load_to_lds %0, %1"
               :: "s"(g0), "s"(g1)
               : "memory");
}

__global__ __launch_bounds__(256)
void contamination_stencil_f32(const float* __restrict__ x,
                               float* __restrict__ y) {
  __shared__ float tile[RROWS][LSTRIDE];

  const int tid = threadIdx.x;
  const int x0  = blockIdx.x * TX;
  const int y0  = blockIdx.y * TY;
  const size_t plane = (size_t)blockIdx.z * (size_t)(HH * WW);
  const float* __restrict__ xp = x + plane;
  float* __restrict__ yp       = y + plane;

  // Interior block: staged region [y0-1, y0+TY] x [x0-4, x0+131] fully in-bounds.
  const bool interior = (blockIdx.x >= 1) && (x0 + TX + 4 <= WW) &&
                        (blockIdx.y >= 1) && (y0 + TY + 1 <= HH);

  if (interior) {
    // Single TDM descriptor moves the whole 34x136 f32 tile (18.5 KB).
    if (tid < 32) {                    // wave 0 issues; EXEC ignored by TDM
      tdm_load_tile(lds_byte_addr(&tile[0][0]),
                    xp + (size_t)(y0 - 1) * WW + (x0 - 4));
      wait_tensorcnt0();               // issuing wave waits for DMA completion
    }
    __syncthreads();                   // publish LDS tile to all 8 waves
  } else {
    // Zero-pad semantics: clear the region, then async-load valid elements.
    f32x4 z = {0.f, 0.f, 0.f, 0.f};
    f32x4* lp = (f32x4*)&tile[0][0];
    for (int i = tid; i < (RROWS * LSTRIDE) / 4; i += 256) lp[i] = z;
    __syncthreads();   // zeros complete before async data can land

    // Needed columns: c in [3,133)  <->  gx in [x0-1, x0+128]
    const int NE = RROWS * 130;                  // 4420
    for (int i = tid; i < NE; i += 256) {
      int r  = i / 130;
      int cc = i - r * 130;
      int c  = cc + 3;
      int gy = y0 - 1 + r;
      int gx = x0 - 1 + cc;
      if ((unsigned)gy < (unsigned)HH && (unsigned)gx < (unsigned)WW) {
        async_load_b32(lds_byte_addr(&tile[r][c]),
                       xp + (size_t)gy * WW + gx);
      }
    }
    wait_asynccnt0();  // this wave's async transfers complete
    __syncthreads();   // all waves' transfers visible
  }

  // Compute: thread -> 4 consecutive x (b128 store) in 4 row-groups.
  // tile[r][c] holds input (gy = y0-1+r, gx = x0-4+c); output (oy,j) uses
  // rows oy..oy+2, cols j+3..j+5.
  const int j0 = (tid & 31) * 4;
  const int rb = tid >> 5;

#pragma unroll
  for (int s = 0; s < 4; ++s) {
    const int oy = rb + s * 8;
    const float* __restrict__ t0 = &tile[oy][0];
    const float* __restrict__ t1 = &tile[oy + 1][0];
    const float* __restrict__ t2 = &tile[oy + 2][0];

    f32x4 o;
#pragma unroll
    for (int k = 0; k < 4; ++k) {
      const int c = j0 + 4 + k;
      float nb = t0[c - 1] + t0[c] + t0[c + 1]
               + t1[c - 1]         + t1[c + 1]
               + t2[c - 1] + t2[c] + t2[c + 1];
      o[k] = t1[c] + ALPHA * nb;
    }
    // Output is write-once: non-temporal store keeps L2 for input halo reuse.
    __builtin_nontemporal_store(
        o, (f32x4*)(yp + (size_t)(y0 + oy) * WW + x0 + j0));
  }
}

extern "C" void kernel_launch(void* const* d_in, const int* in_sizes, int n_in,
                              void* d_out, int out_size, void* d_ws, size_t ws_size,
                              hipStream_t stream) {
  const float* x = (const float*)d_in[0];
  float* y       = (float*)d_out;

  dim3 grid(WW / TX, HH / TY, NPLANES);   // (8, 32, 128)
  dim3 block(256);                        // 8 waves (wave32)
  contamination_stencil_f32<<<grid, block, 0, stream>>>(x, y);
}